// Create_83193516524118
// MI455X (gfx1250) — compile-verified
//
#include <hip/hip_runtime.h>

// CDNA5 / gfx1250 fused GRU + linear projection.
// Uses V_WMMA_F32_16X16X4_F32 (wave32 fp32 matrix core) for all three GEMMs.
// All WMMA code is wave-uniform straight-line code (EXEC all-1s); the r/z vs n
// gate distinction is handled purely in the elementwise epilogue.

typedef float v2f __attribute__((ext_vector_type(2)));
typedef float v8f __attribute__((ext_vector_type(8)));

#define GRU_H   64
#define GRU_I   32
#define GRU_O   16
#define GRU_NG  192      // 3*H
#define BM      16       // batch rows per block (one WMMA M-tile)
#define NWAVES  12       // one 16-wide N-tile of the 192 gate dim per wave
#define WAVE_SZ 32

__global__ __launch_bounds__(NWAVES * WAVE_SZ)
void gru_fused_wmma_kernel(const float* __restrict__ x,      // [B, T, I]
                           const float* __restrict__ W_ih,   // [3H, I]
                           const float* __restrict__ W_hh,   // [3H, H]
                           const float* __restrict__ b_ih,   // [3H]
                           const float* __restrict__ b_hh,   // [3H]
                           const float* __restrict__ W_lin,  // [O, H]
                           const float* __restrict__ b_lin,  // [O]
                           float* __restrict__ out,          // [B, T, O]
                           int T)
{
    __shared__ float h_buf[BM * GRU_H];   // current hidden state (fp32)
    __shared__ float r_buf[BM * GRU_H];   // r gate, written by waves 0-3
    __shared__ float z_buf[BM * GRU_H];   // z gate, written by waves 4-7

    const int tid  = threadIdx.x;
    const int wave = tid / WAVE_SZ;
    const int lane = tid % WAVE_SZ;
    const int hi   = lane >> 4;           // lane half: 0 or 1
    const int ln   = lane & 15;           // 0..15  (M row / N col index)
    const int b0   = blockIdx.x * BM;

    const int  ncol = wave * 16;          // base column in 192-wide gate dim
    const int  gcol = ncol & (GRU_H - 1); // column within a single gate (0..63)
    const bool is_n = (wave >= 8);

    // ---------------- per-wave B fragments (weights), loaded once ----------
    // fp32 4x16 B fragment layout (2 VGPRs): v[j] -> K = 4*kc + 2*hi + j, N = ln
    v2f Bih[GRU_I / 4];     // W_ih^T slice, K=32 -> 8 chunks
    v2f Bhh[GRU_H / 4];     // W_hh^T slice, K=64 -> 16 chunks
    v2f Blin[GRU_H / 4];    // W_lin^T slice (used by wave 0 projection)
    {
        const float* wih_row = W_ih + (size_t)(ncol + ln) * GRU_I;
        #pragma unroll
        for (int kc = 0; kc < GRU_I / 4; ++kc) {
            Bih[kc][0] = wih_row[4 * kc + 2 * hi + 0];
            Bih[kc][1] = wih_row[4 * kc + 2 * hi + 1];
        }
        const float* whh_row = W_hh + (size_t)(ncol + ln) * GRU_H;
        #pragma unroll
        for (int kc = 0; kc < GRU_H / 4; ++kc) {
            Bhh[kc][0] = whh_row[4 * kc + 2 * hi + 0];
            Bhh[kc][1] = whh_row[4 * kc + 2 * hi + 1];
        }
        const float* wl_row = W_lin + (size_t)ln * GRU_H;
        #pragma unroll
        for (int kc = 0; kc < GRU_H / 4; ++kc) {
            Blin[kc][0] = wl_row[4 * kc + 2 * hi + 0];
            Blin[kc][1] = wl_row[4 * kc + 2 * hi + 1];
        }
    }
    const float bih_l  = b_ih[ncol + ln];
    const float bhh_l  = b_hh[ncol + ln];
    const float blin_l = b_lin[ln];

    // h0 = 0
    for (int i = tid; i < BM * GRU_H; i += blockDim.x) h_buf[i] = 0.0f;
    __syncthreads();

    const float* xrow = x + ((size_t)(b0 + ln) * T) * GRU_I;  // this lane's batch row

    for (int t = 0; t < T; ++t) {
        const float* xt = xrow + (size_t)t * GRU_I;
        // speculative prefetch of next timestep's activations (OOB-safe)
        __builtin_prefetch((const void*)(xt + GRU_I), 0, 1);

        // Three independent WMMA accumulation chains (better XDL interleave):
        v8f acc_x  = {0.f, 0.f, 0.f, 0.f, 0.f, 0.f, 0.f, 0.f};  // x_t @ W_ih^T
        v8f acc_h0 = {0.f, 0.f, 0.f, 0.f, 0.f, 0.f, 0.f, 0.f};  // h @ W_hh^T (K 0..31)
        v8f acc_h1 = {0.f, 0.f, 0.f, 0.f, 0.f, 0.f, 0.f, 0.f};  // h @ W_hh^T (K 32..63)

        // ---- gx tile: x_t[16x32] @ W_ih^T[32x16], 8 chained K=4 WMMAs ----
        #pragma unroll
        for (int kc = 0; kc < GRU_I / 4; ++kc) {
            v2f a;
            a[0] = xt[4 * kc + 2 * hi + 0];
            a[1] = xt[4 * kc + 2 * hi + 1];
            acc_x = __builtin_amdgcn_wmma_f32_16x16x4_f32(
                false, a, false, Bih[kc], (short)0, acc_x, false, false);
        }

        // ---- gh tile: h[16x64] @ W_hh^T[64x16], 2x8 chained K=4 WMMAs ----
        #pragma unroll
        for (int kc = 0; kc < 8; ++kc) {
            v2f a;
            a[0] = h_buf[ln * GRU_H + 4 * kc + 2 * hi + 0];
            a[1] = h_buf[ln * GRU_H + 4 * kc + 2 * hi + 1];
            acc_h0 = __builtin_amdgcn_wmma_f32_16x16x4_f32(
                false, a, false, Bhh[kc], (short)0, acc_h0, false, false);
        }
        #pragma unroll
        for (int kc = 8; kc < 16; ++kc) {
            v2f a;
            a[0] = h_buf[ln * GRU_H + 4 * kc + 2 * hi + 0];
            a[1] = h_buf[ln * GRU_H + 4 * kc + 2 * hi + 1];
            acc_h1 = __builtin_amdgcn_wmma_f32_16x16x4_f32(
                false, a, false, Bhh[kc], (short)0, acc_h1, false, false);
        }

        // ---- r / z gates -> LDS (elementwise epilogue, divergence OK) ----
        if (!is_n) {
            #pragma unroll
            for (int v = 0; v < 8; ++v) {
                const int m = v + 8 * hi;
                float g = acc_x[v] + acc_h0[v] + acc_h1[v] + bih_l + bhh_l;
                g = 1.0f / (1.0f + __expf(-g));               // sigmoid
                if (wave < 4) r_buf[m * GRU_H + gcol + ln] = g;
                else          z_buf[m * GRU_H + gcol + ln] = g;
            }
        }
        __syncthreads();

        // ---- n gate + hidden-state update (waves 8-11 own their columns) ----
        if (is_n) {
            #pragma unroll
            for (int v = 0; v < 8; ++v) {
                const int m   = v + 8 * hi;
                const int idx = m * GRU_H + gcol + ln;
                const float r  = r_buf[idx];
                const float z  = z_buf[idx];
                const float gx = acc_x[v] + bih_l;
                const float gh = acc_h0[v] + acc_h1[v] + bhh_l;
                const float n  = tanhf(gx + r * gh);
                const float ho = h_buf[idx];
                h_buf[idx] = (1.0f - z) * n + z * ho;
            }
        }
        __syncthreads();

        // ---- output projection y_t = h_new @ W_lin^T + b_lin (wave 0) ----
        if (wave == 0) {
            v8f acc_y0 = {0.f, 0.f, 0.f, 0.f, 0.f, 0.f, 0.f, 0.f};
            v8f acc_y1 = {0.f, 0.f, 0.f, 0.f, 0.f, 0.f, 0.f, 0.f};
            #pragma unroll
            for (int kc = 0; kc < 8; ++kc) {
                v2f a;
                a[0] = h_buf[ln * GRU_H + 4 * kc + 2 * hi + 0];
                a[1] = h_buf[ln * GRU_H + 4 * kc + 2 * hi + 1];
                acc_y0 = __builtin_amdgcn_wmma_f32_16x16x4_f32(
                    false, a, false, Blin[kc], (short)0, acc_y0, false, false);
            }
            #pragma unroll
            for (int kc = 8; kc < 16; ++kc) {
                v2f a;
                a[0] = h_buf[ln * GRU_H + 4 * kc + 2 * hi + 0];
                a[1] = h_buf[ln * GRU_H + 4 * kc + 2 * hi + 1];
                acc_y1 = __builtin_amdgcn_wmma_f32_16x16x4_f32(
                    false, a, false, Blin[kc], (short)0, acc_y1, false, false);
            }
            #pragma unroll
            for (int v = 0; v < 8; ++v) {
                const int m = v + 8 * hi;
                out[((size_t)(b0 + m) * T + t) * GRU_O + ln] =
                    acc_y0[v] + acc_y1[v] + blin_l;
            }
        }
        // No barrier needed here: h_buf is only rewritten by n-waves after the
        // *next* iteration's first __syncthreads(), which wave 0 reaches only
        // after finishing its projection reads.
    }
}

extern "C" void kernel_launch(void* const* d_in, const int* in_sizes, int n_in,
                              void* d_out, int out_size, void* d_ws, size_t ws_size,
                              hipStream_t stream) {
    const float* x     = (const float*)d_in[0];
    const float* W_ih  = (const float*)d_in[1];
    const float* W_hh  = (const float*)d_in[2];
    const float* b_ih  = (const float*)d_in[3];
    const float* b_hh  = (const float*)d_in[4];
    const float* W_lin = (const float*)d_in[5];
    const float* b_lin = (const float*)d_in[6];
    float* out = (float*)d_out;

    const int B = 512;                                // per reference setup
    const int T = in_sizes[0] / (B * GRU_I);          // 512

    dim3 grid(B / BM);                                // 32 blocks
    dim3 block(NWAVES * WAVE_SZ);                     // 384 threads = 12 wave32s
    gru_fused_wmma_kernel<<<grid, block, 0, stream>>>(
        x, W_ih, W_hh, b_ih, b_hh, W_lin, b_lin, out, T);
}